// PRISM_71511205479155
// MI455X (gfx1250) — compile-verified
//
#include <hip/hip_runtime.h>
#include <hip/hip_bf16.h>
#include <math.h>
#include <stdint.h>

#define B_IMGS 8
#define N_PROP 2048
#define K_OUT  100
#define NTHREADS 1024
#define SCORE_THRESH 0.05f
#define NMS_THRESH   0.4f
#define MIN_SIZE     0.01f
#define IMG_WF 512.0f
#define IMG_HF 512.0f
#define BBOX_CLIP 4.135166556742356f   /* log(1000/16) */

typedef uint32_t u32;
typedef uint64_t u64;
typedef __attribute__((ext_vector_type(4))) u32 u32x4;
typedef __attribute__((ext_vector_type(8))) int i32x8;
typedef __attribute__((ext_vector_type(4))) int i32x4;

// ---------------------------------------------------------------------------
// TDM: DMA one image's proposal tile (2048 x 4 f32 = 32KB, contiguous) from
// global memory into LDS. D# layout per CDNA5 ISA ch.8 (CDNA5 ISA).
// Issued by wave 0 only (TDM ignores EXEC; waves 1..31 branch around on
// execz). Completion tracked with TENSORcnt.
// ---------------------------------------------------------------------------
__device__ __forceinline__ void tdm_load_tile_to_lds(const float* gsrc,
                                                     void* lds_dst,
                                                     u32 n_dwords) {
  u64 ga = (u64)(uintptr_t)gsrc;
  u32 lds_addr = (u32)(uintptr_t)lds_dst;   // low 32 bits = LDS byte offset

  u32x4 g0;
  g0.x = 1u;                                  // count = 1 valid descriptor
  g0.y = lds_addr;                            // lds_addr [63:32]
  g0.z = (u32)ga;                             // global_addr[31:0]
  g0.w = (u32)((ga >> 32) & 0x01FFFFFFu)      // global_addr[56:32]
       | (2u << 30);                          // type = 2 ("image")

  i32x8 g1;
  g1[0] = (int)(2u << 16);                    // data_size = 2 (4 bytes)
  g1[1] = (int)((n_dwords & 0xFFFFu) << 16);  // tensor_dim0[15:0]  @ bits 63:48
  g1[2] = (int)((n_dwords >> 16) | (1u << 16)); // tensor_dim0[31:16] | tensor_dim1=1
  g1[3] = (int)((n_dwords & 0xFFFFu) << 16);  // tile_dim0 @ bits 127:112
  g1[4] = 1;                                  // tile_dim1 = 1, tile_dim2 = 0
  g1[5] = (int)n_dwords;                      // tensor_dim0_stride[31:0]
  g1[6] = 0;                                  // stride hi / dim1_stride lo
  g1[7] = 0;

  i32x4 z4 = {0, 0, 0, 0};
#if __clang_major__ >= 23
  i32x8 z8 = {0, 0, 0, 0, 0, 0, 0, 0};
  __builtin_amdgcn_tensor_load_to_lds(g0, g1, z4, z4, z8, 0);
#else
  __builtin_amdgcn_tensor_load_to_lds(g0, g1, z4, z4, 0);
#endif
}

__global__ __launch_bounds__(NTHREADS)
void PRISM_rcnn_post_kernel(const float* __restrict__ clss,   // [B,N,2]
                            const float* __restrict__ regs,   // [B,N,4]
                            const float* __restrict__ qlts,   // [B,N,1]
                            const float* __restrict__ props,  // [B,N,4]
                            float* __restrict__ out) {
  __shared__ float sbox[N_PROP][4];   // props (TDM in) -> decoded clipped boxes
  __shared__ float skey[N_PROP];      // masked score (-1 if invalid)
  __shared__ int   sidx[N_PROP];      // permutation
  __shared__ int   skeep[N_PROP];     // NMS keep flags
  __shared__ int   smaxbits;          // block max coord (float bits, >=0)

  const int tid  = threadIdx.x;
  const int b    = blockIdx.x;
  const int base = b * N_PROP;

  // ---- Phase 0: async TDM stage of props[b] into LDS; prefetch the rest ----
  if (tid < warpSize) {                       // wave 0 only (wave32)
    tdm_load_tile_to_lds(props + (size_t)base * 4, &sbox[0][0], N_PROP * 4);
  }
  if (tid == 0) smaxbits = 0;
  // pull regs/cls/qlt cachelines toward the WGP while the TDM runs
  __builtin_prefetch(regs + ((size_t)base + tid) * 4, 0, 0);
  __builtin_prefetch(clss + ((size_t)base + tid) * 2, 0, 0);
  __builtin_prefetch(qlts + (size_t)base + tid, 0, 0);
  if (tid < warpSize) {
    __builtin_amdgcn_s_wait_tensorcnt(0);     // TENSORcnt == 0: tile in LDS
  }
  __syncthreads();

  // ---- Phase 1: decode + clip + score, build sort keys ----------------------
  float lmax = 0.0f;
  for (int j = tid; j < N_PROP; j += NTHREADS) {
    float x1 = sbox[j][0], y1 = sbox[j][1];
    float x2 = sbox[j][2], y2 = sbox[j][3];
    const float4 r = reinterpret_cast<const float4*>(regs)[base + j];
    float c0 = clss[(size_t)(base + j) * 2 + 0];
    float c1 = clss[(size_t)(base + j) * 2 + 1];
    float qv = qlts[base + j];

    float w  = x2 - x1,        h  = y2 - y1;
    float cx = x1 + 0.5f * w,  cy = y1 + 0.5f * h;
    float dx = r.x / 10.0f,    dy = r.y / 10.0f;
    float dw = fminf(r.z / 5.0f, BBOX_CLIP);
    float dh = fminf(r.w / 5.0f, BBOX_CLIP);
    float pcx = dx * w + cx,   pcy = dy * h + cy;
    float pw  = expf(dw) * w,  ph  = expf(dh) * h;

    float bx1 = fminf(fmaxf(pcx - 0.5f * pw, 0.0f), IMG_WF);
    float by1 = fminf(fmaxf(pcy - 0.5f * ph, 0.0f), IMG_HF);
    float bx2 = fminf(fmaxf(pcx + 0.5f * pw, 0.0f), IMG_WF);
    float by2 = fminf(fmaxf(pcy + 0.5f * ph, 0.0f), IMG_HF);

    bool small_ok = (bx2 - bx1 >= MIN_SIZE) && (by2 - by1 >= MIN_SIZE);
    float raw   = 1.0f / (1.0f + expf(c0 - c1));   // softmax(...)[:,1]
    float q     = 1.0f / (1.0f + expf(-qv));       // sigmoid(qlt)
    float score = q * raw;
    bool valid  = (raw > SCORE_THRESH) && small_ok;

    sbox[j][0] = bx1; sbox[j][1] = by1; sbox[j][2] = bx2; sbox[j][3] = by2;
    skey[j] = valid ? score : -1.0f;
    sidx[j] = j;
    lmax = fmaxf(lmax, fmaxf(fmaxf(bx1, by1), fmaxf(bx2, by2)));
  }
  atomicMax(&smaxbits, __float_as_int(lmax));      // coords >= 0: int cmp valid
  __syncthreads();
  const float off = __int_as_float(smaxbits) + 1.0f;  // batched_nms class offset

  // ---- Phase 2: bitonic sort (desc by key, stable via index tie-break) ------
  for (int k = 2; k <= N_PROP; k <<= 1) {
    for (int j = k >> 1; j > 0; j >>= 1) {
      for (int t = tid; t < N_PROP; t += NTHREADS) {
        int ixj = t ^ j;
        if (ixj > t) {
          bool up   = ((t & k) == 0);
          float ka = skey[t], kb = skey[ixj];
          int   ia = sidx[t], ib = sidx[ixj];
          bool aPrecedes = (ka > kb) || (ka == kb && ia < ib);
          if (aPrecedes != up) {
            skey[t] = kb; skey[ixj] = ka;
            sidx[t] = ib; sidx[ixj] = ia;
          }
        }
      }
      __syncthreads();
    }
  }

  // ---- Phase 3: sequential greedy NMS, emit top-K inline --------------------
  for (int t = tid; t < N_PROP; t += NTHREADS)
    skeep[t] = (skey[t] >= 0.0f) ? 1 : 0;
  __syncthreads();

  float* ob = out + (size_t)b * K_OUT * 4;
  float* os = out + (size_t)B_IMGS * K_OUT * 4 + (size_t)b * K_OUT;
  float* oc = out + (size_t)B_IMGS * K_OUT * 4 + (size_t)B_IMGS * K_OUT
                  + (size_t)b * K_OUT;

  int count = 0;
  for (int i = 0; i < N_PROP && count < K_OUT; ++i) {
    if (!skeep[i]) continue;                 // workgroup-uniform read
    int ii = sidx[i];
    float ix1 = sbox[ii][0], iy1 = sbox[ii][1];
    float ix2 = sbox[ii][2], iy2 = sbox[ii][3];
    if (tid == 0) {
      ob[count * 4 + 0] = ix1; ob[count * 4 + 1] = iy1;
      ob[count * 4 + 2] = ix2; ob[count * 4 + 3] = iy2;
      os[count] = skey[i];
      oc[count] = 1.0f;                      // single foreground class
    }
    // offset coords exactly as batched_nms does (uniform since C-1 == 1)
    float ox1 = ix1 + off, oy1 = iy1 + off, ox2 = ix2 + off, oy2 = iy2 + off;
    float iarea = (ox2 - ox1) * (oy2 - oy1);
    for (int t = tid; t < N_PROP; t += NTHREADS) {
      if (t > i && skeep[t]) {
        int jj = sidx[t];
        float jx1 = sbox[jj][0] + off, jy1 = sbox[jj][1] + off;
        float jx2 = sbox[jj][2] + off, jy2 = sbox[jj][3] + off;
        float lx = fmaxf(ox1, jx1), ly = fmaxf(oy1, jy1);
        float rx = fminf(ox2, jx2), ry = fminf(oy2, jy2);
        float iw = fmaxf(rx - lx, 0.0f), ih = fmaxf(ry - ly, 0.0f);
        float inter = iw * ih;
        float jarea = (jx2 - jx1) * (jy2 - jy1);
        float iou = inter / (iarea + jarea - inter + 1e-9f);
        if (iou > NMS_THRESH) skeep[t] = 0;
      }
    }
    ++count;
    __syncthreads();                         // make suppressions visible
  }

  // ---- Phase 4: zero-fill unused output slots (reference scatters to zeros) -
  for (int s = count + tid; s < K_OUT; s += NTHREADS) {
    ob[s * 4 + 0] = 0.0f; ob[s * 4 + 1] = 0.0f;
    ob[s * 4 + 2] = 0.0f; ob[s * 4 + 3] = 0.0f;
    os[s] = 0.0f;
    oc[s] = 0.0f;
  }
}

extern "C" void kernel_launch(void* const* d_in, const int* in_sizes, int n_in,
                              void* d_out, int out_size, void* d_ws, size_t ws_size,
                              hipStream_t stream) {
  const float* clss  = (const float*)d_in[0];   // [B,N,2]
  const float* regs  = (const float*)d_in[1];   // [B,N,4]
  const float* qlts  = (const float*)d_in[2];   // [B,N,1]
  const float* props = (const float*)d_in[3];   // [B,N,4]
  float* out = (float*)d_out;                   // boxes | scores | classes
  PRISM_rcnn_post_kernel<<<dim3(B_IMGS), dim3(NTHREADS), 0, stream>>>(
      clss, regs, qlts, props, out);
}